// Decoder_72773925864238
// MI455X (gfx1250) — compile-verified
//
#include <hip/hip_runtime.h>
#include <math.h>

typedef __attribute__((ext_vector_type(2))) float v2f;
typedef __attribute__((ext_vector_type(8))) float v8f;

#define B_     32
#define T_OUT_ 64
#define S_     512
#define H_     512
#define IN_    256
#define V_     32000

// ---------------------------------------------------------------------------
// Generic M=32 GEMM:  C[32 x N] = [A1 | A2] @ W^T + bias
//   A1: 32 x K1 (row stride lda1), A2: 32 x K2 (row stride lda2, may be null)
//   W:  N x (K1+K2) row-major (W row n = column n of B)
// One wave per PAIR of 16-wide N-tiles covering both 16-row M-tiles:
//   per K-step: 2 B-loads + 2 A-loads -> 4 v_wmma_f32_16x16x4_f32
// (A loads amortized across the two N-tiles; W is read exactly once.)
// Requires N % 32 == 0 for the launched shapes (1536, 32000: both ok).
// ---------------------------------------------------------------------------
__global__ void gemm32_wmma(const float* __restrict__ A1, int lda1, int K1,
                            const float* __restrict__ A2, int lda2, int K2,
                            const float* __restrict__ W,  int ldw,
                            const float* __restrict__ bias,
                            float* __restrict__ C, int ldc, int N)
{
    const int wave   = threadIdx.x >> 5;
    const int lane   = threadIdx.x & 31;
    const int ntile0 = (blockIdx.x * (blockDim.x >> 5) + wave) * 2;
    if (ntile0 * 16 >= N) return;              // wave-uniform

    const int l15 = lane & 15;                 // = n for B/D, = m for A
    const int kh  = lane >> 4;                 // 0/1: which K half-pair this lane holds
    const int gn0 = ntile0 * 16 + l15;
    const int gn1 = gn0 + 16;

    const float* wrow0 = W + (size_t)gn0 * ldw + 2 * kh;
    const float* wrow1 = W + (size_t)gn1 * ldw + 2 * kh;
    const float* a1r0  = A1 + (size_t)l15 * lda1 + 2 * kh;
    const float* a1r1  = A1 + (size_t)(l15 + 16) * lda1 + 2 * kh;

    v8f acc00 = {};   // (mtile0, ntile0)
    v8f acc10 = {};   // (mtile1, ntile0)
    v8f acc01 = {};   // (mtile0, ntile1)
    v8f acc11 = {};   // (mtile1, ntile1)

    for (int kk = 0; kk < K1; kk += 4) {
        float2 bv0 = *(const float2*)(wrow0 + kk);
        float2 bv1 = *(const float2*)(wrow1 + kk);
        float2 av0 = *(const float2*)(a1r0 + kk);
        float2 av1 = *(const float2*)(a1r1 + kk);
        v2f b0; b0[0] = bv0.x; b0[1] = bv0.y;
        v2f b1; b1[0] = bv1.x; b1[1] = bv1.y;
        v2f a0; a0[0] = av0.x; a0[1] = av0.y;
        v2f a1; a1[0] = av1.x; a1[1] = av1.y;
        acc00 = __builtin_amdgcn_wmma_f32_16x16x4_f32(false, a0, false, b0, (short)0, acc00, false, false);
        acc10 = __builtin_amdgcn_wmma_f32_16x16x4_f32(false, a1, false, b0, (short)0, acc10, false, false);
        acc01 = __builtin_amdgcn_wmma_f32_16x16x4_f32(false, a0, false, b1, (short)0, acc01, false, false);
        acc11 = __builtin_amdgcn_wmma_f32_16x16x4_f32(false, a1, false, b1, (short)0, acc11, false, false);
    }
    if (A2) {
        const float* w0k   = wrow0 + K1;
        const float* w1k   = wrow1 + K1;
        const float* a2r0  = A2 + (size_t)l15 * lda2 + 2 * kh;
        const float* a2r1  = A2 + (size_t)(l15 + 16) * lda2 + 2 * kh;
        for (int kk = 0; kk < K2; kk += 4) {
            float2 bv0 = *(const float2*)(w0k + kk);
            float2 bv1 = *(const float2*)(w1k + kk);
            float2 av0 = *(const float2*)(a2r0 + kk);
            float2 av1 = *(const float2*)(a2r1 + kk);
            v2f b0; b0[0] = bv0.x; b0[1] = bv0.y;
            v2f b1; b1[0] = bv1.x; b1[1] = bv1.y;
            v2f a0; a0[0] = av0.x; a0[1] = av0.y;
            v2f a1; a1[0] = av1.x; a1[1] = av1.y;
            acc00 = __builtin_amdgcn_wmma_f32_16x16x4_f32(false, a0, false, b0, (short)0, acc00, false, false);
            acc10 = __builtin_amdgcn_wmma_f32_16x16x4_f32(false, a1, false, b0, (short)0, acc10, false, false);
            acc01 = __builtin_amdgcn_wmma_f32_16x16x4_f32(false, a0, false, b1, (short)0, acc01, false, false);
            acc11 = __builtin_amdgcn_wmma_f32_16x16x4_f32(false, a1, false, b1, (short)0, acc11, false, false);
        }
    }

    const float bb0 = bias ? bias[gn0] : 0.0f;
    const float bb1 = bias ? bias[gn1] : 0.0f;
    const int mbase = kh * 8;                  // D layout: lane half selects M 0-7 vs 8-15
    #pragma unroll
    for (int i = 0; i < 8; ++i) {
        C[(size_t)(mbase + i)      * ldc + gn0] = acc00[i] + bb0;
        C[(size_t)(mbase + i + 16) * ldc + gn0] = acc10[i] + bb0;
        C[(size_t)(mbase + i)      * ldc + gn1] = acc01[i] + bb1;
        C[(size_t)(mbase + i + 16) * ldc + gn1] = acc11[i] + bb1;
    }
}

// ---------------------------------------------------------------------------
// enc_logit[b,s] = hiddens[b,s,:] . att_W[0, H:]
// ---------------------------------------------------------------------------
__global__ void enc_logit_kernel(const float* __restrict__ hiddens,
                                 const float* __restrict__ aWe,
                                 float* __restrict__ enc)
{
    __shared__ float red[128];
    const int row = blockIdx.x;                // b*S + s
    const int tid = threadIdx.x;
    const float* hr = hiddens + (size_t)row * H_;
    float p = 0.f;
    for (int k = tid; k < H_; k += 128) p += hr[k] * aWe[k];
    red[tid] = p; __syncthreads();
    for (int s = 64; s > 0; s >>= 1) {
        if (tid < s) red[tid] += red[tid + s];
        __syncthreads();
    }
    if (tid == 0) enc[row] = red[0];
}

// ---------------------------------------------------------------------------
// Attention: e = relu(mask*(h1.aW_h) + enc + att_b); softmax over S; c = attn@hiddens
// One block per batch row, 256 threads (2 positions / 2 H-columns per thread).
// ---------------------------------------------------------------------------
__global__ void attn_kernel(const float* __restrict__ h1,
                            const float* __restrict__ hiddens,
                            const float* __restrict__ enc,
                            const float* __restrict__ att_W,
                            const float* __restrict__ att_b,
                            const int*   __restrict__ lengths,
                            float* __restrict__ cctx)
{
    __shared__ float red[256];
    __shared__ float attn_s[S_];
    const int b = blockIdx.x, tid = threadIdx.x;

    // sval = h1[b] . aW_h
    const float* hr = h1 + b * H_;
    float p = 0.f;
    for (int k = tid; k < H_; k += 256) p += hr[k] * att_W[k];
    red[tid] = p; __syncthreads();
    for (int s = 128; s > 0; s >>= 1) { if (tid < s) red[tid] += red[tid + s]; __syncthreads(); }
    const float sval = red[0]; __syncthreads();

    const float ab  = att_b[0];
    const int   len = lengths[b];
    const float* er = enc + b * S_;
    const int s0 = tid, s1 = tid + 256;
    const float m0 = (s0 < len) ? 1.f : 0.f;
    const float m1 = (s1 < len) ? 1.f : 0.f;
    const float e0 = fmaxf(m0 * sval + er[s0] + ab, 0.f);
    const float e1 = fmaxf(m1 * sval + er[s1] + ab, 0.f);

    red[tid] = fmaxf(e0, e1); __syncthreads();
    for (int s = 128; s > 0; s >>= 1) { if (tid < s) red[tid] = fmaxf(red[tid], red[tid + s]); __syncthreads(); }
    const float mx = red[0]; __syncthreads();

    const float x0 = expf(e0 - mx), x1 = expf(e1 - mx);
    red[tid] = x0 + x1; __syncthreads();
    for (int s = 128; s > 0; s >>= 1) { if (tid < s) red[tid] += red[tid + s]; __syncthreads(); }
    const float inv = 1.0f / red[0]; __syncthreads();

    attn_s[s0] = x0 * inv;
    attn_s[s1] = x1 * inv;
    __syncthreads();

    // c[b, tid] and c[b, tid+256]
    float a0 = 0.f, a1 = 0.f;
    const float* hb = hiddens + (size_t)b * S_ * H_;
    for (int s = 0; s < S_; ++s) {
        const float w = attn_s[s];
        const float* row = hb + (size_t)s * H_;
        a0 += w * row[tid];
        a1 += w * row[tid + 256];
    }
    cctx[b * H_ + tid]       = a0;
    cctx[b * H_ + tid + 256] = a1;
}

// ---------------------------------------------------------------------------
// GRU gate fusion: r = sig(gi_r+gh_r); z = sig(gi_z+gh_z); n = tanh(gi_n + r*gh_n)
// h = (1-z)*n + z*h      (gi/gh are 32 x 1536, h is 32 x 512)
// ---------------------------------------------------------------------------
__global__ void gru_gate_kernel(const float* __restrict__ gi,
                                const float* __restrict__ gh,
                                float* __restrict__ h)
{
    const int idx = blockIdx.x * blockDim.x + threadIdx.x;   // 32*512 total
    const int b = idx >> 9, j = idx & (H_ - 1);
    const float* gib = gi + (size_t)b * 3 * H_;
    const float* ghb = gh + (size_t)b * 3 * H_;
    const float r = 1.f / (1.f + expf(-(gib[j]          + ghb[j])));
    const float z = 1.f / (1.f + expf(-(gib[j + H_]     + ghb[j + H_])));
    const float n = tanhf(gib[j + 2 * H_] + r * ghb[j + 2 * H_]);
    h[idx] = (1.f - z) * n + z * h[idx];
}

// ---------------------------------------------------------------------------
// Per-row max + log-sum-exp over V=32000
// ---------------------------------------------------------------------------
__global__ void lse_kernel(const float* __restrict__ logits, float* __restrict__ stats)
{
    __shared__ float red[256];
    const int b = blockIdx.x, tid = threadIdx.x;
    const float* row = logits + (size_t)b * V_;
    float mx = -3.4e38f;
    for (int v = tid; v < V_; v += 256) mx = fmaxf(mx, row[v]);
    red[tid] = mx; __syncthreads();
    for (int s = 128; s > 0; s >>= 1) { if (tid < s) red[tid] = fmaxf(red[tid], red[tid + s]); __syncthreads(); }
    mx = red[0]; __syncthreads();
    float sum = 0.f;
    for (int v = tid; v < V_; v += 256) sum += expf(row[v] - mx);
    red[tid] = sum; __syncthreads();
    for (int s = 128; s > 0; s >>= 1) { if (tid < s) red[tid] += red[tid + s]; __syncthreads(); }
    if (tid == 0) { stats[2 * b] = mx; stats[2 * b + 1] = logf(red[0]); }
}

__global__ void out_write_kernel(const float* __restrict__ logits,
                                 const float* __restrict__ stats,
                                 float* __restrict__ out, int t)
{
    const int idx = blockIdx.x * blockDim.x + threadIdx.x;   // 32 * 32000
    const int b = idx / V_, v = idx - b * V_;
    out[(size_t)b * T_OUT_ * V_ + (size_t)t * V_ + v] =
        logits[idx] - stats[2 * b] - stats[2 * b + 1];
}

// ---------------------------------------------------------------------------
extern "C" void kernel_launch(void* const* d_in, const int* in_sizes, int n_in,
                              void* d_out, int out_size, void* d_ws, size_t ws_size,
                              hipStream_t stream)
{
    const float* inputs       = (const float*)d_in[0];
    const int*   lengths      = (const int*)  d_in[1];
    const float* final_hidden = (const float*)d_in[2];
    const float* hiddens      = (const float*)d_in[3];
    const float* att_W        = (const float*)d_in[4];
    const float* att_b        = (const float*)d_in[5];
    const float* out_W        = (const float*)d_in[6];
    const float* out_b        = (const float*)d_in[7];
    const float* W_ih0        = (const float*)d_in[8];
    const float* W_hh0        = (const float*)d_in[9];
    const float* b_ih0        = (const float*)d_in[10];
    const float* b_hh0        = (const float*)d_in[11];
    const float* W_ih1        = (const float*)d_in[12];
    const float* W_hh1        = (const float*)d_in[13];
    const float* b_ih1        = (const float*)d_in[14];
    const float* b_hh1        = (const float*)d_in[15];
    float* out = (float*)d_out;

    // workspace layout (floats)
    float* ws     = (float*)d_ws;
    float* h0     = ws;                       // 32*512
    float* h1     = ws + 16384;               // 32*512
    float* cctx   = ws + 32768;               // 32*512
    float* enc    = ws + 49152;               // 32*512
    float* gi     = ws + 65536;               // 32*1536
    float* gh     = ws + 65536 + 49152;       // 32*1536
    float* stats  = ws + 65536 + 98304;       // 64
    float* logits = stats + 64;               // 32*32000

    // init recurrent state: final_hidden is (L=2, B, H) contiguous -> h0 then h1
    hipMemcpyAsync(h0, final_hidden, (size_t)2 * B_ * H_ * sizeof(float),
                   hipMemcpyDeviceToDevice, stream);

    // enc_logit = hiddens @ att_W[0, H:]  (once)
    enc_logit_kernel<<<B_ * S_, 128, 0, stream>>>(hiddens, att_W + H_, enc);

    for (int t = 0; t < T_OUT_; ++t) {
        // attention context from previous h1
        attn_kernel<<<B_, 256, 0, stream>>>(h1, hiddens, enc, att_W, att_b,
                                            lengths, cctx);

        // --- GRU layer 0: x0 = [x_t (256) | c (512)] ---
        // 1536/16 = 96 tiles -> 48 waves -> 12 blocks of 4 waves
        gemm32_wmma<<<12, 128, 0, stream>>>(inputs + (size_t)t * IN_, T_OUT_ * IN_, IN_,
                                            cctx, H_, H_,
                                            W_ih0, IN_ + H_, b_ih0,
                                            gi, 3 * H_, 3 * H_);
        gemm32_wmma<<<12, 128, 0, stream>>>(h0, H_, H_,
                                            nullptr, 0, 0,
                                            W_hh0, H_, b_hh0,
                                            gh, 3 * H_, 3 * H_);
        gru_gate_kernel<<<64, 256, 0, stream>>>(gi, gh, h0);

        // --- GRU layer 1 ---
        gemm32_wmma<<<12, 128, 0, stream>>>(h0, H_, H_,
                                            nullptr, 0, 0,
                                            W_ih1, H_, b_ih1,
                                            gi, 3 * H_, 3 * H_);
        gemm32_wmma<<<12, 128, 0, stream>>>(h1, H_, H_,
                                            nullptr, 0, 0,
                                            W_hh1, H_, b_hh1,
                                            gh, 3 * H_, 3 * H_);
        gru_gate_kernel<<<64, 256, 0, stream>>>(gi, gh, h1);

        // --- vocab projection: feat = [h1 | c] @ out_W^T + out_b ---
        // 32000/16 = 2000 tiles -> 1000 waves -> 250 blocks of 4 waves
        gemm32_wmma<<<250, 128, 0, stream>>>(h1, H_, H_,
                                             cctx, H_, H_,
                                             out_W, 2 * H_, out_b,
                                             logits, V_, V_);

        // --- log_softmax ---
        lse_kernel<<<B_, 256, 0, stream>>>(logits, stats);
        out_write_kernel<<<(B_ * V_) / 256, 256, 0, stream>>>(logits, stats, out, t);
    }
}